// TreeLSTMCellv2_25254407701045
// MI455X (gfx1250) — compile-verified
//
#include <hip/hip_runtime.h>

typedef __attribute__((ext_vector_type(2))) float v2f;
typedef __attribute__((ext_vector_type(8))) float v8f;

#define HS 128      // H_SIZE
#define XS 256      // X_SIZE
#define GS 512      // 4*H
#define XP 260      // LDS pitch for x tile (pad 4 floats -> conflict-free)
#define HP 132      // LDS pitch for h/c tiles

// ---------------------------------------------------------------- zero scratch
__global__ __launch_bounds__(256) void zero_ws_kernel(float4* __restrict__ ws, int n4) {
    int i = blockIdx.x * 256 + threadIdx.x;
    if (i < n4) ws[i] = float4{0.f, 0.f, 0.f, 0.f};
}

// ------------------------------------------------- edge scatter (segment sum)
// one wave32 per edge; lane owns 4 of 128 columns -> float4 gather + 8 fp32 atomics
__global__ __launch_bounds__(256) void scatter_hc_kernel(
    const float* __restrict__ h, const float* __restrict__ c,
    const int* __restrict__ src, const int* __restrict__ dst,
    float* __restrict__ ws_h, float* __restrict__ ws_c, int E) {
    int e = blockIdx.x * 8 + (threadIdx.x >> 5);
    if (e >= E) return;
    int lane = threadIdx.x & 31;
    int s = src[e];
    int d = dst[e];
    int col = lane * 4;
    const float4 hv = *(const float4*)(h + (size_t)s * HS + col);
    const float4 cv = *(const float4*)(c + (size_t)s * HS + col);
    float* ph = ws_h + (size_t)d * HS + col;
    float* pc = ws_c + (size_t)d * HS + col;
    __hip_atomic_fetch_add(ph + 0, hv.x, __ATOMIC_RELAXED, __HIP_MEMORY_SCOPE_AGENT);
    __hip_atomic_fetch_add(ph + 1, hv.y, __ATOMIC_RELAXED, __HIP_MEMORY_SCOPE_AGENT);
    __hip_atomic_fetch_add(ph + 2, hv.z, __ATOMIC_RELAXED, __HIP_MEMORY_SCOPE_AGENT);
    __hip_atomic_fetch_add(ph + 3, hv.w, __ATOMIC_RELAXED, __HIP_MEMORY_SCOPE_AGENT);
    __hip_atomic_fetch_add(pc + 0, cv.x, __ATOMIC_RELAXED, __HIP_MEMORY_SCOPE_AGENT);
    __hip_atomic_fetch_add(pc + 1, cv.y, __ATOMIC_RELAXED, __HIP_MEMORY_SCOPE_AGENT);
    __hip_atomic_fetch_add(pc + 2, cv.z, __ATOMIC_RELAXED, __HIP_MEMORY_SCOPE_AGENT);
    __hip_atomic_fetch_add(pc + 3, cv.w, __ATOMIC_RELAXED, __HIP_MEMORY_SCOPE_AGENT);
}

__device__ __forceinline__ float fast_sigmoid(float g) {
    return 1.0f / (1.0f + __expf(-g));      // exp->inf => 0 ; exp->0 => 1 (safe)
}
__device__ __forceinline__ float fast_tanh(float g) {
    return 1.0f - 2.0f / (1.0f + __expf(2.0f * g));  // safe at +/-inf
}

// ------------------------------------- fused dual-GEMM (f32 WMMA) + LSTM gates
// block = 256 threads = 8 waves, one 16-node tile per block.
// wave w owns output columns [w*16, w*16+16) and the matching i/o/u/f gate
// column tiles at +0/+128/+256/+384 of the 512-wide preactivation matrix, so
// gates are evaluated entirely in WMMA accumulator registers.
__global__ __launch_bounds__(256) void fused_wmma_gates_kernel(
    const float* __restrict__ x,
    const float* __restrict__ W,    // [256,512] row-major
    const float* __restrict__ bW,   // [512]
    const float* __restrict__ U,    // [128,512] row-major
    const float* __restrict__ bU,   // [512]
    const float* __restrict__ ws_h, // [N,128] aggregated h_in
    const float* __restrict__ ws_c, // [N,128] aggregated c_in
    float* __restrict__ out_h,      // [N,128]
    float* __restrict__ out_c,      // [N,128]
    int n_nodes) {

    __shared__ float lx[16 * XP];
    __shared__ float lh[16 * HP];
    __shared__ float lc[16 * HP];

    const int tid   = threadIdx.x;
    const int wave  = tid >> 5;
    const int lane  = tid & 31;
    const int node0 = blockIdx.x * 16;

    // ---- stage x tile [16 x 256] coalesced (float4), row clamped for tail
#pragma unroll
    for (int it = 0; it < 4; ++it) {
        int idx = tid + it * 256;           // 0..1023
        int row = idx >> 6;                 // /64 float4 per row
        int k4  = idx & 63;
        int rg  = node0 + row; if (rg >= n_nodes) rg = n_nodes - 1;
        float4 v = *(const float4*)(x + (size_t)rg * XS + k4 * 4);
        *(float4*)&lx[row * XP + k4 * 4] = v;
    }
    // ---- stage h_in / c_in tiles [16 x 128]
#pragma unroll
    for (int it = 0; it < 2; ++it) {
        int idx = tid + it * 256;           // 0..511
        int row = idx >> 5;                 // /32 float4 per row
        int k4  = idx & 31;
        int rg  = node0 + row; if (rg >= n_nodes) rg = n_nodes - 1;
        float4 vh = *(const float4*)(ws_h + (size_t)rg * HS + k4 * 4);
        float4 vc = *(const float4*)(ws_c + (size_t)rg * HS + k4 * 4);
        *(float4*)&lh[row * HP + k4 * 4] = vh;
        *(float4*)&lc[row * HP + k4 * 4] = vc;
    }
    __syncthreads();

    const int m16   = lane & 15;            // A-row / C-col index within tile
    const int khalf = (lane >> 4) * 2;      // K sub-offset: lanes 16-31 hold K+2
    const int jcol  = wave * 16 + m16;      // this lane's output column (0..127)

    v8f acc_i = {}, acc_o = {}, acc_u = {}, acc_f = {};

    // ---- g += x @ W   (K = 256, step 4, V_WMMA_F32_16X16X4_F32)
#pragma unroll 4
    for (int k0 = 0; k0 < XS; k0 += 4) {
        v2f a = *(const v2f*)&lx[m16 * XP + k0 + khalf];      // ds_load_b64
        const float* wr0 = W + (size_t)(k0 + khalf) * GS + jcol;
        const float* wr1 = wr0 + GS;
        v2f b;
        b.x = wr0[0];   b.y = wr1[0];
        acc_i = __builtin_amdgcn_wmma_f32_16x16x4_f32(false, a, false, b, (short)0, acc_i, false, false);
        b.x = wr0[128]; b.y = wr1[128];
        acc_o = __builtin_amdgcn_wmma_f32_16x16x4_f32(false, a, false, b, (short)0, acc_o, false, false);
        b.x = wr0[256]; b.y = wr1[256];
        acc_u = __builtin_amdgcn_wmma_f32_16x16x4_f32(false, a, false, b, (short)0, acc_u, false, false);
        b.x = wr0[384]; b.y = wr1[384];
        acc_f = __builtin_amdgcn_wmma_f32_16x16x4_f32(false, a, false, b, (short)0, acc_f, false, false);
    }

    // ---- g += h_in @ U   (K = 128)
#pragma unroll 4
    for (int k0 = 0; k0 < HS; k0 += 4) {
        v2f a = *(const v2f*)&lh[m16 * HP + k0 + khalf];
        const float* ur0 = U + (size_t)(k0 + khalf) * GS + jcol;
        const float* ur1 = ur0 + GS;
        v2f b;
        b.x = ur0[0];   b.y = ur1[0];
        acc_i = __builtin_amdgcn_wmma_f32_16x16x4_f32(false, a, false, b, (short)0, acc_i, false, false);
        b.x = ur0[128]; b.y = ur1[128];
        acc_o = __builtin_amdgcn_wmma_f32_16x16x4_f32(false, a, false, b, (short)0, acc_o, false, false);
        b.x = ur0[256]; b.y = ur1[256];
        acc_u = __builtin_amdgcn_wmma_f32_16x16x4_f32(false, a, false, b, (short)0, acc_u, false, false);
        b.x = ur0[384]; b.y = ur1[384];
        acc_f = __builtin_amdgcn_wmma_f32_16x16x4_f32(false, a, false, b, (short)0, acc_f, false, false);
    }

    // ---- combined biases per gate for this lane's column
    const float bi = bW[jcol]       + bU[jcol];
    const float bo = bW[128 + jcol] + bU[128 + jcol];
    const float bu = bW[256 + jcol] + bU[256 + jcol];
    const float bf = bW[384 + jcol] + bU[384 + jcol];

    // ---- gates + state update straight from accumulators
    // D-matrix layout: VGPR r holds row r (lanes 0-15) / row r+8 (lanes 16-31)
    const int mbase = (lane >> 4) << 3;
#pragma unroll
    for (int r = 0; r < 8; ++r) {
        int m = r + mbase;
        int n = node0 + m;
        if (n >= n_nodes) continue;
        float ig = fast_sigmoid(acc_i[r] + bi);
        float og = fast_sigmoid(acc_o[r] + bo);
        float ug = fast_tanh   (acc_u[r] + bu);
        float fg = fast_sigmoid(acc_f[r] + bf);
        float cin  = lc[m * HP + jcol];
        float cnew = ig * ug + fg * cin;
        float hnew = og * fast_tanh(cnew);
        size_t off = (size_t)n * HS + jcol;
        out_h[off] = hnew;
        out_c[off] = cnew;
    }
}

// ---------------------------------------------------------------------- launch
extern "C" void kernel_launch(void* const* d_in, const int* in_sizes, int n_in,
                              void* d_out, int out_size, void* d_ws, size_t ws_size,
                              hipStream_t stream) {
    const float* x   = (const float*)d_in[0];
    const float* h   = (const float*)d_in[1];
    const float* c   = (const float*)d_in[2];
    const int*   src = (const int*)d_in[3];
    const int*   dst = (const int*)d_in[4];
    const float* W   = (const float*)d_in[5];
    const float* bW  = (const float*)d_in[6];
    const float* U   = (const float*)d_in[7];
    const float* bU  = (const float*)d_in[8];

    const int N = in_sizes[1] / HS;   // 200000
    const int E = in_sizes[3];        // 500000

    float* ws_h = (float*)d_ws;
    float* ws_c = ws_h + (size_t)N * HS;
    float* out_h = (float*)d_out;
    float* out_c = out_h + (size_t)N * HS;

    // 1) zero mailboxes
    int n4 = (2 * N * HS) / 4;
    zero_ws_kernel<<<(n4 + 255) / 256, 256, 0, stream>>>((float4*)d_ws, n4);

    // 2) edge scatter (segment sums of child h, c)
    scatter_hc_kernel<<<(E + 7) / 8, 256, 0, stream>>>(h, c, src, dst, ws_h, ws_c, E);

    // 3) fused dual-GEMM (WMMA f32) + gates
    int nblk = (N + 15) / 16;
    fused_wmma_gates_kernel<<<nblk, 256, 0, stream>>>(x, W, bW, U, bU, ws_h, ws_c,
                                                      out_h, out_c, N);
}